// CTCLoss_32959579030260
// MI455X (gfx1250) — compile-verified
//
#include <hip/hip_runtime.h>
#include <stdint.h>

// Problem constants (from the reference): T=4096, B=64, P=512, F=5, SHARP=1.
#define T_STEPS   4096
#define BATCH     64
#define P_POS     512
#define F_FEAT    5
#define CHUNK     512                    // time steps per TDM chunk
#define NCHUNK    (T_STEPS / CHUNK)
#define PHASES    (CHUNK / 2)            // 2 time steps per barrier phase
#define NEG_LARGE (-1e30f)

typedef unsigned int v4u __attribute__((ext_vector_type(4)));
typedef int          v8i __attribute__((ext_vector_type(8)));
typedef int          v4i __attribute__((ext_vector_type(4)));

// Issue one TDM tensor_load_to_lds: a 2D tile of CHUNK rows x 5 f32,
// row stride = B*F elements, into LDS at byte offset lds_off.
// D# packing per CDNA5 ISA ch.8 (groups 0/1; groups 2/3 zero => 2D tensor).
__device__ __forceinline__ void tdm_load_chunk(const float* gsrc, uint32_t lds_off) {
  uint64_t ga = (uint64_t)(uintptr_t)gsrc;
  v4u g0;
  g0.x = 1u;                                              // count=1 (valid descriptor)
  g0.y = lds_off;                                         // lds_addr (bytes)
  g0.z = (uint32_t)(ga & 0xFFFFFFFFull);                  // global_addr[31:0]
  g0.w = (uint32_t)((ga >> 32) & 0x01FFFFFFull)           // global_addr[56:32]
       | (2u << 30);                                      // type=2 ("image")
  v8i g1;
  g1[0] = (int)(2u << 16);                                // data_size=2 (4 bytes)
  g1[1] = (int)(F_FEAT << 16);                            // tensor_dim0 = 5 (bits 79:48)
  g1[2] = (int)(CHUNK << 16);                             // tensor_dim1 = 512 (bits 111:80)
  g1[3] = (int)(F_FEAT << 16);                            // tile_dim0 = 5 (bits 127:112)
  g1[4] = (int)CHUNK;                                     // tile_dim1 = 512, tile_dim2 = 0
  g1[5] = (int)(BATCH * F_FEAT);                          // tensor_dim0_stride = 320
  g1[6] = 0;                                              // stride hi / dim1_stride lo
  g1[7] = 0;                                              // tensor_dim1_stride hi
  v4i z4 = {0, 0, 0, 0};
  v8i z8 = {0, 0, 0, 0, 0, 0, 0, 0};
  // 6-arg toolchain signature: (g0, g1, g2, g3, <v8i>, cpol)
  __builtin_amdgcn_tensor_load_to_lds(g0, g1, z4, z4, z8, /*cpol=*/0);
}

// logaddexp(a,b) = max + log(1 + exp(min-max)); exact for -1e30 sentinels
// (exp of a huge negative underflows to 0, giving max back exactly).
__device__ __forceinline__ float logaddexp_f(float a, float b) {
  float m = fmaxf(a, b);
  float d = fminf(a, b) - m;                              // <= 0
  return m + __logf(1.0f + __expf(d));
}

__global__ __launch_bounds__(P_POS) void ctc_fwd_scan_kernel(
    const float* __restrict__ x,        // [T, B, F]
    const int*   __restrict__ seqs,     // [B, P]
    const int*   __restrict__ seqlens,  // [B]
    float*       __restrict__ out)      // [B, 1]
{
  // Double-buffered emission stripe: 2 * 512 steps * 5 f32 = 20 KB.
  __shared__ float xbuf[2 * CHUNK * F_FEAT];
  // Double-buffered forward scores, shifted by 1:
  //   gbuf[buf][0]   = -1e30 sentinel (position -1, never written)
  //   gbuf[buf][q+1] = F[q] for q = 0..512
  __shared__ float gbuf[2][P_POS + 2];

  const int tid = threadIdx.x;          // owns position p = tid+1
  const int b   = blockIdx.x;           // one workgroup per batch element

  // Emission channels: ca feeds moves into p (uses seqs[p-1] = seqs[tid]);
  // cb feeds moves into p-1 (uses seqs[tid-1]; clamped for tid==0 — the
  // -1e30 halo sentinel makes its contribution vanish regardless).
  const int ca = seqs[(size_t)b * P_POS + tid];
  const int cb = seqs[(size_t)b * P_POS + (tid > 0 ? tid - 1 : 0)];
  const int sl = seqlens[b];

  // Initial state: F[0] = 0, F[1..512] = -1e30, sentinel at index 0.
  gbuf[0][tid + 2] = NEG_LARGE;
  if (tid == 0) {
    gbuf[0][0] = NEG_LARGE;  gbuf[1][0] = NEG_LARGE;      // position -1 sentinels
    gbuf[0][1] = 0.0f;                                    // F[0]
  }
  float f  = NEG_LARGE;                 // F[tid+1], register resident
  float f0 = 0.0f;                      // F[0] (meaningful on thread 0 only)

  const float* xb = x + (size_t)b * F_FEAT;               // &x[0][b][0]
  const uint32_t lds_base = (uint32_t)(uintptr_t)&xbuf[0];

  // Prime the pipeline: DMA chunk 0 into buffer 0 (wave 0 only; TDM ignores
  // EXEC, the other waves branch around the call entirely).
  if (tid < 32) {
    tdm_load_chunk(xb, lds_base);
  }

  int cur = 0;
  for (int c = 0; c < NCHUNK; ++c) {
    if (tid < 32) {
      __builtin_amdgcn_s_wait_tensorcnt(0);               // chunk c landed in LDS
      if (c + 1 < NCHUNK) {                               // prefetch chunk c+1
        const float* gsrc = xb + (size_t)(c + 1) * CHUNK * BATCH * F_FEAT;
        tdm_load_chunk(gsrc, lds_base + (uint32_t)(((c + 1) & 1) * (CHUNK * F_FEAT * 4)));
      }
    }
    __syncthreads();                                      // publish chunk c to all waves

    const float* xc = &xbuf[(c & 1) * (CHUNK * F_FEAT)];
    #pragma unroll 2
    for (int ph = 0; ph < PHASES; ++ph) {
      const int lt = ph * 2;                              // local time within chunk
      // Emissions for both steps of this phase (LDS, broadcast-friendly).
      float es0 = xc[lt * F_FEAT + 4];                    // stay, step t
      float ea0 = xc[lt * F_FEAT + ca];                   // move into p, step t
      float eb0 = xc[lt * F_FEAT + cb];                   // move into p-1, step t
      float es1 = xc[(lt + 1) * F_FEAT + 4];              // stay, step t+1
      float ea1 = xc[(lt + 1) * F_FEAT + ca];             // move into p, step t+1

      float h1 = gbuf[cur][tid + 1];                      // F_t[p-1]
      float h0 = gbuf[cur][tid];                          // F_t[p-2] (-1e30 at edge)

      // Step t:   own position + redundant left-neighbor update.
      float fA  = logaddexp_f(ea0 + h1, es0 + f);         // F_{t+1}[p]
      float h1A = logaddexp_f(eb0 + h0, es0 + h1);        // F_{t+1}[p-1]
      // Step t+1: own position only.
      f = logaddexp_f(ea1 + h1A, es1 + fA);               // F_{t+2}[p]

      int nxt = cur ^ 1;
      gbuf[nxt][tid + 2] = f;
      f0 += es0 + es1;                                    // branchless F[0] track
      if (tid == 0) gbuf[nxt][1] = f0;
      __syncthreads();                                    // one barrier per 2 steps
      cur = nxt;
    }
  }

  const float scale = -1.0f / (float)T_STEPS;             // SHARP == 1
  if (tid + 1 == sl)        out[b] = f  * scale;
  if (tid == 0 && sl == 0)  out[b] = f0 * scale;
}

extern "C" void kernel_launch(void* const* d_in, const int* in_sizes, int n_in,
                              void* d_out, int out_size, void* d_ws, size_t ws_size,
                              hipStream_t stream) {
  const float* x       = (const float*)d_in[0];   // [4096, 64, 5] f32
  const int*   seqs    = (const int*)d_in[1];     // [64, 512] i32
  const int*   seqlens = (const int*)d_in[2];     // [64] i32
  float*       out     = (float*)d_out;           // [64, 1] f32

  ctc_fwd_scan_kernel<<<BATCH, P_POS, 0, stream>>>(x, seqs, seqlens, out);
}